// SyntheticRoutePropagationNetwork_36043365548105
// MI455X (gfx1250) — compile-verified
//
#include <hip/hip_runtime.h>

typedef __attribute__((ext_vector_type(2))) float v2f;
typedef __attribute__((ext_vector_type(8))) float v8f;

constexpr int kNodes   = 50000;
constexpr int kProp    = 25000;
constexpr int kEdges   = 600000;
constexpr int kHidden  = 128;
constexpr int kMTiles  = (kProp + 15) / 16;   // 1563
constexpr int kPadRows = kMTiles * 16;        // 25008

// ---------------------------------------------------------------- zero fill
__global__ __launch_bounds__(256) void zero_f4(float4* __restrict__ p, int n4) {
  int i = blockIdx.x * blockDim.x + threadIdx.x;
  if (i < n4) p[i] = make_float4(0.f, 0.f, 0.f, 0.f);
}

// ------------------------------------------------- edge gather + scatter-add
// One wave32 per edge: each lane gathers float4 (16B) of the 512B source row
// and issues 4 hardware f32 atomics (global_atomic_add_f32; dst region fits
// in the 192MB L2 so these resolve in-cache).
__global__ __launch_bounds__(256) void scatter_add_rows(
    const float* __restrict__ feat, const int* __restrict__ src,
    const int* __restrict__ dst, float* __restrict__ acc, int n_edges) {
  int wave = (blockIdx.x * blockDim.x + threadIdx.x) >> 5;
  int lane = threadIdx.x & 31;
  if (wave >= n_edges) return;
  int s = src[wave];
  int d = dst[wave];
  float4 v = ((const float4*)(feat + (size_t)s * kHidden))[lane];
  float* p = acc + (size_t)d * kHidden + (lane << 2);
  unsafeAtomicAdd(p + 0, v.x);
  unsafeAtomicAdd(p + 1, v.y);
  unsafeAtomicAdd(p + 2, v.z);
  unsafeAtomicAdd(p + 3, v.w);
}

// ------------------------------------- WMMA f32 GEMM + bias + relu + add-out
// out[r,:] += relu(A[r,:] @ W^T + b)  for r < kProp.
// One wave per 16-row M tile. A tile (16x128 f32) kept in 32 v2f fragments;
// for each of 8 N tiles, 32 chained v_wmma_f32_16x16x4_f32 accumulate K=128.
__global__ __launch_bounds__(256) void gemm_relu_add(
    const float* __restrict__ A,     // [kPadRows, 128] zero-padded parent_agg
    const float* __restrict__ W,     // [128, 128] row-major (out, in)
    const float* __restrict__ bias,  // [128]
    float* __restrict__ out) {       // [kNodes, 128]
  int wave = (blockIdx.x * blockDim.x + threadIdx.x) >> 5;
  int lane = threadIdx.x & 31;
  if (wave >= kMTiles) return;       // uniform per wave: EXEC stays all-1s

  const int half = lane >> 4;        // selects K-pair {0,1} vs {2,3} per step
  const int lid  = lane & 15;        // A: row-in-tile m; B/C/D: column n

  // A fragments: afrag[ks] = A[tile_row + lid][4*ks + 2*half .. +1]
  const v2f* a2 = (const v2f*)(A + ((size_t)wave * 16 + lid) * kHidden);
  v2f afrag[32];
#pragma unroll
  for (int ks = 0; ks < 32; ++ks) afrag[ks] = a2[2 * ks + half];

  for (int nt = 0; nt < 8; ++nt) {
    const int n = nt * 16 + lid;                       // output column
    const v2f* w2 = (const v2f*)(W + (size_t)n * kHidden); // B[k][n] = W[n][k]
    v8f acc = {};
#pragma unroll
    for (int ks = 0; ks < 32; ++ks) {
      v2f bfrag = w2[2 * ks + half];
      acc = __builtin_amdgcn_wmma_f32_16x16x4_f32(
          /*neg_a=*/false, afrag[ks], /*neg_b=*/false, bfrag,
          /*c_mod=*/(short)0, acc, /*reuse_a=*/false, /*reuse_b=*/false);
    }
    const float bv = bias[n];
    const int mbase = wave * 16 + half * 8;            // C/D: VGPR r -> row mbase+r
#pragma unroll
    for (int r = 0; r < 8; ++r) {
      const int row = mbase + r;
      if (row < kProp) {
        float* o = out + (size_t)row * kHidden + n;
        *o += fmaxf(acc[r] + bv, 0.0f);
      }
    }
  }
}

// --------------------------------------------------------------------- host
extern "C" void kernel_launch(void* const* d_in, const int* in_sizes, int n_in,
                              void* d_out, int out_size, void* d_ws, size_t ws_size,
                              hipStream_t stream) {
  const float* prop_z      = (const float*)d_in[0];
  const float* mol_z       = (const float*)d_in[1];
  const float* W           = (const float*)d_in[2];
  const float* b           = (const float*)d_in[3];
  const int*   parent_src  = (const int*)d_in[4];
  const int*   parent_dst  = (const int*)d_in[5];
  const int*   sibling_src = (const int*)d_in[6];
  const int*   sibling_dst = (const int*)d_in[7];
  float* out        = (float*)d_out;
  float* parent_agg = (float*)d_ws;  // [kPadRows, 128] f32 = 12.8 MB

  // 1) out = prop_z
  hipMemcpyAsync(out, prop_z, (size_t)kNodes * kHidden * sizeof(float),
                 hipMemcpyDeviceToDevice, stream);

  // 2) parent_agg = 0 (padded rows included so GEMM reads are in-bounds)
  const int n4 = kPadRows * kHidden / 4;
  zero_f4<<<(n4 + 255) / 256, 256, 0, stream>>>((float4*)parent_agg, n4);

  // 3) parent_agg[dst] += prop_z[src]   (1 wave per edge, 8 edges/block)
  scatter_add_rows<<<kEdges / 8, 256, 0, stream>>>(
      prop_z, parent_src, parent_dst, parent_agg, kEdges);

  // 4) out[:kProp] += relu(parent_agg @ W^T + b)   (WMMA f32 16x16x4)
  gemm_relu_add<<<(kMTiles + 7) / 8, 256, 0, stream>>>(parent_agg, W, b, out);

  // 5) out[dst] += mol_z[src]   (dst < kProp by construction)
  scatter_add_rows<<<kEdges / 8, 256, 0, stream>>>(
      mol_z, sibling_src, sibling_dst, out, kEdges);
}